// Voxelization_15436112462068
// MI455X (gfx1250) — compile-verified
//
#include <hip/hip_runtime.h>

typedef __attribute__((ext_vector_type(2))) float v2f;
typedef __attribute__((ext_vector_type(8))) float v8f;

#define BB 16
#define CC 64
#define NN 65536
#define RR 32
#define R3 32768
#define GRID_ELEMS (BB * CC * R3)   /* 33554432 */
#define NC_OFF     GRID_ELEMS
#define WS_MEAN 0                    /* 48 floats: per (b,dim) coordinate SUM */
#define WS_MAX  48                   /* 16 floats: per b max L2 norm          */
#define WS_CNT  64                   /* B*R3 floats: voxel counts             */
#define INV_NN (1.0f / 65536.0f)

// ---------------------------------------------------------------- zero fills
__global__ void vox_zero_out(float4* __restrict__ g) {
  unsigned i = blockIdx.x * 256u + threadIdx.x;       // exactly GRID_ELEMS/4 threads
  g[i] = make_float4(0.f, 0.f, 0.f, 0.f);
}

__global__ void vox_zero_ws(float* __restrict__ ws) {
  unsigned i = blockIdx.x * 256u + threadIdx.x;
  if (i < (unsigned)(WS_CNT + BB * R3)) ws[i] = 0.f;
}

// ------------------------------------------------- per-batch coordinate sums
// One block per (b,dim) pair (48 blocks), 4 wave32 per block.
// Each wave chains V_WMMA_F32_16X16X4_F32 with an all-ones B matrix:
//   D[m][n] = sum_k A[m][k] + C[m][n]  -> 16 row accumulators in f32.
// Summing all 16 rows at the end yields the exact (reassociated) f32 sum of
// every element fed through A, so the lane->element mapping is free.
__global__ void vox_mean_wmma(const float* __restrict__ coords,
                              float* __restrict__ ws) {
  const int bd   = blockIdx.x;             // 0..47  -> (b*3 + dim)
  const int lane = threadIdx.x & 31;
  const int wave = threadIdx.x >> 5;       // 0..3
  const float* p = coords + ((size_t)bd << 16) + ((size_t)wave << 14); // 16384/wave

  v8f acc = {};
  v2f ones; ones.x = 1.0f; ones.y = 1.0f;  // B = all ones (4x16)

  for (int i = 0; i < 256; ++i) {          // 256 iters * 64 elems = 16384
    v2f a;
    a.x = p[lane];
    a.y = p[lane + 32];
    acc = __builtin_amdgcn_wmma_f32_16x16x4_f32(
        /*neg_a=*/false, a, /*neg_b=*/false, ones,
        /*c_mod=*/(short)0, acc, /*reuse_a=*/false, /*reuse_b=*/false);
    p += 64;
  }

  // lane L holds rows 0-7 (L<16) or 8-15 (L>=16) of one (identical) column.
  float s = acc[0] + acc[1] + acc[2] + acc[3] + acc[4] + acc[5] + acc[6] + acc[7];
  float o = __shfl_down(s, 16, 32);
  if (lane == 0) atomicAdd(&ws[WS_MEAN + bd], s + o);   // 4 waves accumulate
}

// ------------------------------------------------------- per-batch max norm
__global__ void vox_maxnorm(const float* __restrict__ coords,
                            float* __restrict__ ws) {
  const int b = blockIdx.x;
  const float mx = ws[WS_MEAN + b * 3 + 0] * INV_NN;
  const float my = ws[WS_MEAN + b * 3 + 1] * INV_NN;
  const float mz = ws[WS_MEAN + b * 3 + 2] * INV_NN;
  const float* cb = coords + ((size_t)(b * 3) << 16);

  float vmax = 0.0f;
  for (int n = threadIdx.x; n < NN; n += 256) {
    float dx = cb[n] - mx;
    float dy = cb[NN + n] - my;
    float dz = cb[2 * NN + n] - mz;
    float d2 = dx * dx + dy * dy + dz * dz;
    vmax = fmaxf(vmax, d2);
  }
  __shared__ float red[256];
  red[threadIdx.x] = vmax;
  __syncthreads();
  for (int off = 128; off > 0; off >>= 1) {
    if (threadIdx.x < (unsigned)off)
      red[threadIdx.x] = fmaxf(red[threadIdx.x], red[threadIdx.x + off]);
    __syncthreads();
  }
  if (threadIdx.x == 0) ws[WS_MAX + b] = sqrtf(red[0]);  // sqrt(max d2) == max ||.||
}

// ------------------------------------------------------------------ scatter
// One thread per point: writes clipped nc, scatter-adds count + 64 channels.
__global__ void vox_scatter(const float* __restrict__ features,
                            const float* __restrict__ coords,
                            float* __restrict__ ws,
                            float* __restrict__ out) {
  const unsigned t = blockIdx.x * 256u + threadIdx.x;   // 0 .. B*N-1
  const int b = t >> 16;
  const int n = t & (NN - 1);

  const float mx = ws[WS_MEAN + b * 3 + 0] * INV_NN;
  const float my = ws[WS_MEAN + b * 3 + 1] * INV_NN;
  const float mz = ws[WS_MEAN + b * 3 + 2] * INV_NN;
  const float m  = ws[WS_MAX + b];
  const float sc = 16.0f / m;   // ((c-mean)/(2m) + 0.5) * 32 == (c-mean)*(16/m) + 16

  const float* cb = coords + ((size_t)(b * 3) << 16);
  float x = (cb[n]          - mx) * sc + 16.0f;
  float y = (cb[NN + n]     - my) * sc + 16.0f;
  float z = (cb[2 * NN + n] - mz) * sc + 16.0f;
  x = fminf(fmaxf(x, 0.0f), 31.0f);
  y = fminf(fmaxf(y, 0.0f), 31.0f);
  z = fminf(fmaxf(z, 0.0f), 31.0f);

  float* ncout = out + NC_OFF + ((size_t)(b * 3) << 16);
  ncout[n]          = x;
  ncout[NN + n]     = y;
  ncout[2 * NN + n] = z;

  const int vx = (int)rintf(x);  // round-half-even, matches jnp.round
  const int vy = (int)rintf(y);
  const int vz = (int)rintf(z);
  const int idx = (vx * RR + vy) * RR + vz;

  atomicAdd(&ws[WS_CNT + (b << 15) + idx], 1.0f);

  const float* fb = features + ((size_t)b << 22) + n;    // b*C*N + n
  float*       gb = out + ((size_t)b << 21) + idx;       // b*C*R3 + idx
#pragma unroll 4
  for (int c = 0; c < CC; ++c) {
    __builtin_prefetch(fb + ((size_t)(c + 4) << 16), 0, 0);
    atomicAdd(gb + ((size_t)c << 15), fb[(size_t)c << 16]);
  }
}

// ----------------------------------------------------------------- finalize
__global__ void vox_finalize(float* __restrict__ out, const float* __restrict__ ws) {
  const unsigned e = blockIdx.x * 256u + threadIdx.x;    // 0 .. GRID_ELEMS-1
  const unsigned b = e >> 21;                            // / (C*R3)
  const unsigned v = e & (R3 - 1);
  const float cnt = ws[WS_CNT + (b << 15) + v];
  out[e] = out[e] / fmaxf(cnt, 1.0f);
}

extern "C" void kernel_launch(void* const* d_in, const int* in_sizes, int n_in,
                              void* d_out, int out_size, void* d_ws, size_t ws_size,
                              hipStream_t stream) {
  const float* features = (const float*)d_in[0];   // [16,64,65536] f32
  const float* coords   = (const float*)d_in[1];   // [16,3,65536]  f32
  float* out = (float*)d_out;                      // grid (33554432) ++ nc (3145728)
  float* ws  = (float*)d_ws;                       // 64 + B*R3 floats (~2.1 MB)

  vox_zero_out<<<GRID_ELEMS / 4 / 256, 256, 0, stream>>>((float4*)out);
  vox_zero_ws <<<(WS_CNT + BB * R3 + 255) / 256, 256, 0, stream>>>(ws);
  vox_mean_wmma<<<BB * 3, 128, 0, stream>>>(coords, ws);
  vox_maxnorm  <<<BB, 256, 0, stream>>>(coords, ws);
  vox_scatter  <<<(BB * NN) / 256, 256, 0, stream>>>(features, coords, ws, out);
  vox_finalize <<<GRID_ELEMS / 256, 256, 0, stream>>>(out, ws);
}